// TuckERT_29446295781673
// MI455X (gfx1250) — compile-verified
//
#include <hip/hip_runtime.h>
#include <cstddef>

#define B_SZ  2048
#define DIM   200
#define NENT  100000
#define EPSV  1e-5f

typedef __attribute__((ext_vector_type(2))) float v2f;
typedef __attribute__((ext_vector_type(8))) float v8f;

// ---------------- helpers ----------------
__device__ __forceinline__ float blockReduceSum(float v, float* red) {
    int tid = threadIdx.x;
    red[tid] = v;
    __syncthreads();
    for (int s = blockDim.x >> 1; s > 0; s >>= 1) {
        if (tid < s) red[tid] += red[tid + s];
        __syncthreads();
    }
    float r = red[0];
    __syncthreads();
    return r;
}

// ---------------- kernels ----------------
__global__ void k_zero(float* p, int n) {
    int i = blockIdx.x * 256 + threadIdx.x;
    if (i < n) p[i] = 0.0f;
}

// Gather lhs, rel_t; accumulate Σ(rel_t^4), Σ(R_noT[x1]^4), Σ(E[x2]^4).
__global__ void k_gather(const int* __restrict__ x, const float* __restrict__ E,
                         const float* __restrict__ R, const float* __restrict__ Rn,
                         const float* __restrict__ T,
                         float* __restrict__ lhs, float* __restrict__ relt,
                         float* __restrict__ regacc) {
    __shared__ float red[256];
    int b = blockIdx.x, d = threadIdx.x;
    int i0 = x[b * 4 + 0], i1 = x[b * 4 + 1], i2 = x[b * 4 + 2], i3 = x[b * 4 + 3];
    float rt = 0.0f, rn = 0.0f, rh = 0.0f;
    if (d < DIM) {
        float lv = E[(size_t)i0 * DIM + d];
        float rv = R[i1 * DIM + d];
        float tv = T[i3 * DIM + d];
        rt = rv * tv;
        rn = Rn[i1 * DIM + d];
        rh = E[(size_t)i2 * DIM + d];
        lhs[b * DIM + d]  = lv;
        relt[b * DIM + d] = rt;
    }
    float s1 = blockReduceSum(rt * rt * rt * rt, red);
    float s2 = blockReduceSum(rn * rn * rn * rn, red);
    float s3 = blockReduceSum(rh * rh * rh * rh, red);
    if (threadIdx.x == 0) {
        atomicAdd(&regacc[1], s1);
        atomicAdd(&regacc[2], s2);
        atomicAdd(&regacc[3], s3);
    }
}

// Σ W^4 over 8M elements.
__global__ void k_norm4w(const float* __restrict__ W, float* regacc) {
    __shared__ float red[256];
    size_t i = (size_t)blockIdx.x * 1024 + threadIdx.x;
    float s = 0.0f;
    for (int j = 0; j < 4; ++j) {
        size_t idx = i + (size_t)j * 256;
        if (idx < 8000000ull) { float v = W[idx]; s += v * v * v * v; }
    }
    s = blockReduceSum(s, red);
    if (threadIdx.x == 0) atomicAdd(&regacc[4], s);
}

// BatchNorm stats: one block per feature dim, reduce over batch.
__global__ void k_bnstats(const float* __restrict__ X, float* mean, float* inv) {
    __shared__ float red[256];
    int d = blockIdx.x;
    float s = 0.0f, s2 = 0.0f;
    for (int b = threadIdx.x; b < B_SZ; b += 256) {
        float v = X[b * DIM + d];
        s += v;
        s2 += v * v;
    }
    s  = blockReduceSum(s,  red);
    s2 = blockReduceSum(s2, red);
    if (threadIdx.x == 0) {
        float mu  = s * (1.0f / B_SZ);
        float var = s2 * (1.0f / B_SZ) - mu * mu;
        mean[d] = mu;
        inv[d]  = rsqrtf(var + EPSV);
    }
}

// y = (x-mu)*invstd*gamma+beta; optional Σ y^4 accumulation.
__global__ void k_bnapply(const float* __restrict__ X, const float* __restrict__ mean,
                          const float* __restrict__ inv, const float* __restrict__ gamma,
                          const float* __restrict__ beta, float* __restrict__ Y,
                          float* regdst) {
    __shared__ float red[256];
    int b = blockIdx.x, d = threadIdx.x;
    float y = 0.0f;
    if (d < DIM) {
        y = (X[b * DIM + d] - mean[d]) * inv[d] * gamma[d] + beta[d];
        Y[b * DIM + d] = y;
    }
    if (regdst) {
        float s = blockReduceSum(y * y * y * y, red);
        if (threadIdx.x == 0) atomicAdd(regdst, s);
    }
}

// Core fused GEMM: h[b,e] = Σ_{k,d} relt[b,k]*lhsn[b,d]*W[k,d,e]
// A[b, k*200+d] = relt[b,k]*lhsn[b,d] synthesized from LDS (K = 40000),
// B streamed from W via unconditional loads (overlapping last e-tile, no guards).
// Block = 128 threads (4 waves), tile 32 b-rows x 64 e-cols.
// Wave w: cols [e0+16w, e0+16w+16), rows 0..31 as two 16x16 accumulators
// sharing each B fragment (halves global B loads per WMMA).
__global__ __launch_bounds__(128)
void k_core_gemm(const float* __restrict__ lhsn, const float* __restrict__ relt,
                 const float* __restrict__ W, float* __restrict__ h) {
    __shared__ float ls_l[32 * 204];   // padded stride 204 to spread LDS banks
    __shared__ float ls_r[32 * 204];
    int blk = blockIdx.x;
    int b0 = (blk >> 2) * 32;
    int et = blk & 3;
    int e0 = et * 64;
    if (e0 > DIM - 64) e0 = DIM - 64;          // last tile overlaps: 0,64,128,136
    int tid = threadIdx.x;
    for (int idx = tid; idx < 32 * DIM; idx += 128) {
        int row = idx / DIM, d = idx % DIM;
        ls_l[row * 204 + d] = lhsn[(b0 + row) * DIM + d];
        ls_r[row * 204 + d] = relt[(b0 + row) * DIM + d];
    }
    __syncthreads();

    int w = tid >> 5, lane = tid & 31;
    int half = lane >> 4;                      // K={0,1} vs K={2,3} lane group
    int l15  = lane & 15;
    int lrow0 = l15;                           // accumulator 0: rows 0..15
    int lrow1 = 16 + l15;                      // accumulator 1: rows 16..31
    int ecol  = e0 + w * 16 + l15;             // B/C column (always < 200)
    int doff  = half * 2;
    const float* lp0 = &ls_l[lrow0 * 204];
    const float* lp1 = &ls_l[lrow1 * 204];

    v8f c0 = {}, c1 = {};
    for (int k = 0; k < DIM; ++k) {
        float rk0 = ls_r[lrow0 * 204 + k];
        float rk1 = ls_r[lrow1 * 204 + k];
        const float* wp = W + ((size_t)k * DIM) * DIM + (size_t)doff * DIM + ecol;
#pragma unroll 5
        for (int d4 = 0; d4 < DIM / 4; ++d4) {
            int dbase = d4 * 4 + doff;
            v2f a0, a1, bb;
            bb.x = wp[(size_t)d4 * 4 * DIM];
            bb.y = wp[(size_t)d4 * 4 * DIM + DIM];
            a0.x = rk0 * lp0[dbase];
            a0.y = rk0 * lp0[dbase + 1];
            a1.x = rk1 * lp1[dbase];
            a1.y = rk1 * lp1[dbase + 1];
            c0 = __builtin_amdgcn_wmma_f32_16x16x4_f32(false, a0, false, bb,
                                                       (short)0, c0, false, false);
            c1 = __builtin_amdgcn_wmma_f32_16x16x4_f32(false, a1, false, bb,
                                                       (short)0, c1, false, false);
        }
    }
    int rb0 = b0 + half * 8;                   // C layout: vgpr r -> M = r + 8*half
    int rb1 = b0 + 16 + half * 8;
#pragma unroll
    for (int r = 0; r < 8; ++r) {
        h[(rb0 + r) * DIM + ecol] = c0[r];
        h[(rb1 + r) * DIM + ecol] = c1[r];
    }
}

// pred = h_bn @ E^T : M=2048, N=100000, K=200 via f32 WMMA 16x16x4.
// Block = 128 threads (4 waves), tile 64 b-rows x 32 n-cols.
// Wave w: cols subtile (w&1), rows 32*(w>>1)..+31 as two accumulators sharing
// each 8-byte B fragment loaded from E.
__global__ __launch_bounds__(128)
void k_pred_gemm(const float* __restrict__ hb, const float* __restrict__ E,
                 float* __restrict__ pred) {
    __shared__ float hs[64 * 204];
    const int NT = NENT / 32;                  // 3125 n-tiles
    int blk = blockIdx.x;
    int b0 = (blk / NT) * 64;
    int n0 = (blk % NT) * 32;
    int tid = threadIdx.x;
    for (int idx = tid; idx < 64 * DIM; idx += 128) {
        int row = idx / DIM, d = idx % DIM;
        hs[row * 204 + d] = hb[(b0 + row) * DIM + d];
    }
    __syncthreads();

    int w = tid >> 5, lane = tid & 31;
    int wn = w & 1, wb = w >> 1;
    int half = lane >> 4;
    int l15  = lane & 15;
    int lrow0 = wb * 32 + l15;
    int lrow1 = lrow0 + 16;
    int ncol  = n0 + wn * 16 + l15;
    const float* ep = E + (size_t)ncol * DIM + half * 2;  // 8B-aligned (even offset)
    const float* ap0 = &hs[lrow0 * 204 + half * 2];
    const float* ap1 = &hs[lrow1 * 204 + half * 2];

    v8f c0 = {}, c1 = {};
#pragma unroll 5
    for (int s = 0; s < DIM / 4; ++s) {
        v2f a0, a1, bb;
        float2 bv = *(const float2*)(ep + 4 * s);
        bb.x = bv.x;
        bb.y = bv.y;
        a0.x = ap0[4 * s];
        a0.y = ap0[4 * s + 1];
        a1.x = ap1[4 * s];
        a1.y = ap1[4 * s + 1];
        c0 = __builtin_amdgcn_wmma_f32_16x16x4_f32(false, a0, false, bb,
                                                   (short)0, c0, false, false);
        c1 = __builtin_amdgcn_wmma_f32_16x16x4_f32(false, a1, false, bb,
                                                   (short)0, c1, false, false);
    }
    int rb0 = b0 + wb * 32 + half * 8;
    int rb1 = rb0 + 16;
#pragma unroll
    for (int r = 0; r < 8; ++r) {
        pred[(size_t)(rb0 + r) * NENT + ncol] = c0[r];
        pred[(size_t)(rb1 + r) * NENT + ncol] = c1[r];
    }
}

__global__ void k_finalize(const float* __restrict__ regacc, float* __restrict__ out_reg) {
    int i = threadIdx.x;
    if (i < 5) out_reg[i] = powf(regacc[i], 0.25f);
}

__global__ void k_copyT(const float* __restrict__ T, float* __restrict__ dst, int n) {
    int i = blockIdx.x * 256 + threadIdx.x;
    if (i < n) dst[i] = T[i];
}

// ---------------- launcher ----------------
extern "C" void kernel_launch(void* const* d_in, const int* in_sizes, int n_in,
                              void* d_out, int out_size, void* d_ws, size_t ws_size,
                              hipStream_t stream) {
    const int*   x  = (const int*)  d_in[0];
    const float* E  = (const float*)d_in[1];
    const float* R  = (const float*)d_in[2];
    const float* Rn = (const float*)d_in[3];
    const float* T  = (const float*)d_in[4];
    const float* W  = (const float*)d_in[5];
    const float* g0 = (const float*)d_in[6];
    const float* be0= (const float*)d_in[7];
    const float* g1 = (const float*)d_in[8];
    const float* be1= (const float*)d_in[9];
    float* out = (float*)d_out;
    float* ws  = (float*)d_ws;

    // workspace layout (floats); total ~8.2 MB
    float* lhs    = ws;
    float* relt   = lhs  + (size_t)B_SZ * DIM;
    float* lhsn   = relt + (size_t)B_SZ * DIM;
    float* h      = lhsn + (size_t)B_SZ * DIM;
    float* hb     = h    + (size_t)B_SZ * DIM;
    float* mean0  = hb   + (size_t)B_SZ * DIM;
    float* inv0   = mean0 + 256;
    float* mean1  = inv0  + 256;
    float* inv1   = mean1 + 256;
    float* regacc = inv1  + 256;

    float* pred = out;
    float* rego = out + (size_t)B_SZ * NENT;
    float* To   = rego + 5;

    k_zero<<<1, 256, 0, stream>>>(regacc, 8);
    k_gather<<<B_SZ, 256, 0, stream>>>(x, E, R, Rn, T, lhs, relt, regacc);
    k_norm4w<<<7813, 256, 0, stream>>>(W, regacc);
    k_bnstats<<<DIM, 256, 0, stream>>>(lhs, mean0, inv0);
    k_bnapply<<<B_SZ, 256, 0, stream>>>(lhs, mean0, inv0, g0, be0, lhsn, &regacc[0]);
    k_core_gemm<<<(B_SZ / 32) * 4, 128, 0, stream>>>(lhsn, relt, W, h);
    k_bnstats<<<DIM, 256, 0, stream>>>(h, mean1, inv1);
    k_bnapply<<<B_SZ, 256, 0, stream>>>(h, mean1, inv1, g1, be1, hb, nullptr);
    k_pred_gemm<<<(B_SZ / 64) * (NENT / 32), 128, 0, stream>>>(hb, E, pred);
    k_finalize<<<1, 32, 0, stream>>>(regacc, rego);
    k_copyT<<<(80000 + 255) / 256, 256, 0, stream>>>(T, To, 80000);
}